// MultiHeadAttentionEQ_37477884625166
// MI455X (gfx1250) — compile-verified
//
#include <hip/hip_runtime.h>
#include <math.h>

typedef __attribute__((ext_vector_type(2))) float v2f;
typedef __attribute__((ext_vector_type(8))) float v8f;

#define WMMA_F32(a, b, c) \
  __builtin_amdgcn_wmma_f32_16x16x4_f32(false, (a), false, (b), (short)0, (c), false, false)

// Problem constants
#define A_ 4
#define N_ 384
#define D_ 256
#define H_ 8
#define C_ 32
#define NPAIR 16         // A_*A_ unique (a,e) attention problems
#define NTILE (N_ / 16)  // 24
#define NBLK (H_ * NTILE)            // 192 attention blocks per pair
#define SCALE 0.17677669529663689f   // 1/sqrt(32)

// Workspace layout (in floats)
#define WS_QP   0u
#define WS_KP   393216u
#define WS_VP   786432u
#define WS_O    1179648u             // NPAIR*H_*N_*C_ = 1572864 floats
#define WS_PART 2752512u             // NPAIR*NBLK = 3072 floats
#define WS_W    2755584u             // 16 floats

// ---------------------------------------------------------------------------
// Projection: y = x @ W^T + b, stored head-major as out[a][h][n][c].
// One wave per 16x16 output tile; K=256 swept in chunks of 4 via f32 WMMA.
// ---------------------------------------------------------------------------
__global__ __launch_bounds__(32) void proj_kernel(const float* __restrict__ x,
                                                  const float* __restrict__ W,
                                                  const float* __restrict__ bias,
                                                  float* __restrict__ out) {
  const int lane = threadIdx.x;
  const int lo = lane & 15;
  const int hi = lane >> 4;  // 0 or 1
  const int t0 = blockIdx.x * 16;  // token tile (a*384+n), 96 tiles
  const int d0 = blockIdx.y * 16;  // output-channel tile, 16 tiles

  const float* __restrict__ xrow = x + (size_t)(t0 + lo) * D_;
  const float* __restrict__ wrow = W + (size_t)(d0 + lo) * D_;

  v8f acc = {};
#pragma unroll 8
  for (int kk = 0; kk < D_; kk += 4) {
    v2f av, bv;
    av.x = xrow[kk + 2 * hi];
    av.y = xrow[kk + 2 * hi + 1];
    bv.x = wrow[kk + 2 * hi];
    bv.y = wrow[kk + 2 * hi + 1];
    acc = WMMA_F32(av, bv, acc);
  }

  const float bval = bias[d0 + lo];
  const int d = d0 + lo;
  const int h = d >> 5;
  const int c = d & 31;
  // a-index is uniform per block (tiles never straddle an a boundary):
  const int ai = blockIdx.x / NTILE;          // scalar (SALU) divide
  const int nbase = t0 - ai * N_ + 8 * hi;    // row n for VGPR j is nbase + j
  float* __restrict__ obase = out + (((size_t)(ai * H_ + h)) * N_ + nbase) * C_ + c;
#pragma unroll
  for (int j = 0; j < 8; ++j) {
    obase[(size_t)j * C_] = acc[j] + bval;
  }
}

// ---------------------------------------------------------------------------
// Fused flash attention for one (a,e,h,n-tile): S = (Q*scale) @ K^T via WMMA,
// online softmax, O += P @ V via WMMA. Writes its raw score sum to a private
// partial slot (deterministic reduction later; no float atomics).
// ---------------------------------------------------------------------------
__global__ __launch_bounds__(32) void attn_kernel(const float* __restrict__ qp,
                                                  const float* __restrict__ kp,
                                                  const float* __restrict__ vp,
                                                  float* __restrict__ Obuf,
                                                  float* __restrict__ partial) {
  __shared__ float ps[16][17];  // P transpose staging (C-layout -> A-layout)

  const int lane = threadIdx.x;
  const int lo = lane & 15;
  const int hi = lane >> 4;

  const int n0 = blockIdx.x * 16;
  const int h = blockIdx.y;
  const int pair = blockIdx.z;
  const int a = pair >> 2;
  const int e = pair & 3;

  const float* __restrict__ qb = qp + ((size_t)(a * H_ + h)) * N_ * C_;  // [n][c]
  const float* __restrict__ kb = kp + ((size_t)(e * H_ + h)) * N_ * C_;  // [m][c]
  const float* __restrict__ vb = vp + ((size_t)(e * H_ + h)) * N_ * C_;  // [m][c]

  // Q A-fragments for this n-tile (scale folded in): 8 chunks of K(=c)=4
  v2f qa[8];
#pragma unroll
  for (int c4 = 0; c4 < 8; ++c4) {
    const float* qr = qb + (size_t)(n0 + lo) * C_ + c4 * 4 + 2 * hi;
    qa[c4].x = qr[0] * SCALE;
    qa[c4].y = qr[1] * SCALE;
  }

  v8f o0 = {}, o1 = {};
  float mrow[8], lrow[8];
#pragma unroll
  for (int j = 0; j < 8; ++j) { mrow[j] = -3.0e38f; lrow[j] = 0.0f; }
  float ssum = 0.0f;

  for (int m0 = 0; m0 < N_; m0 += 16) {
    // ---- S tile = (Q*scale) @ K^T : 8 WMMAs over c ----
    v8f s = {};
#pragma unroll
    for (int c4 = 0; c4 < 8; ++c4) {
      const float* kr = kb + (size_t)(m0 + lo) * C_ + c4 * 4 + 2 * hi;
      v2f kv;
      kv.x = kr[0];
      kv.y = kr[1];
      s = WMMA_F32(qa[c4], kv, s);
    }

    // ---- raw score sum (for pooled) + online softmax ----
    v8f p;
#pragma unroll
    for (int j = 0; j < 8; ++j) {
      ssum += s[j];
      float rmax = s[j];
      rmax = fmaxf(rmax, __shfl_xor(rmax, 1));
      rmax = fmaxf(rmax, __shfl_xor(rmax, 2));
      rmax = fmaxf(rmax, __shfl_xor(rmax, 4));
      rmax = fmaxf(rmax, __shfl_xor(rmax, 8));
      const float mnew = fmaxf(mrow[j], rmax);
      const float alpha = __expf(mrow[j] - mnew);
      p[j] = __expf(s[j] - mnew);
      float rsum = p[j];
      rsum += __shfl_xor(rsum, 1);
      rsum += __shfl_xor(rsum, 2);
      rsum += __shfl_xor(rsum, 4);
      rsum += __shfl_xor(rsum, 8);
      lrow[j] = lrow[j] * alpha + rsum;
      mrow[j] = mnew;
      o0[j] *= alpha;
      o1[j] *= alpha;
    }

    // ---- transpose P through LDS: C-layout -> A-layout ----
    __syncthreads();
#pragma unroll
    for (int j = 0; j < 8; ++j) ps[j + 8 * hi][lo] = p[j];
    __syncthreads();

    // ---- O += P @ V : K(=m)=16 in 4 chunks, two c-halves ----
#pragma unroll
    for (int m4 = 0; m4 < 4; ++m4) {
      v2f pa;
      pa.x = ps[lo][m4 * 4 + 2 * hi];
      pa.y = ps[lo][m4 * 4 + 2 * hi + 1];
      const int mg = m0 + m4 * 4 + 2 * hi;
      v2f vb0, vb1;
      vb0.x = vb[(size_t)mg * C_ + lo];
      vb0.y = vb[(size_t)(mg + 1) * C_ + lo];
      vb1.x = vb[(size_t)mg * C_ + 16 + lo];
      vb1.y = vb[(size_t)(mg + 1) * C_ + 16 + lo];
      o0 = WMMA_F32(pa, vb0, o0);
      o1 = WMMA_F32(pa, vb1, o1);
    }
  }

  // ---- normalize and store O[a,e][h][n][c] ----
  const size_t obase = ((size_t)(pair * H_ + h)) * N_ * C_;
#pragma unroll
  for (int j = 0; j < 8; ++j) {
    const float inv = 1.0f / lrow[j];
    const int n = n0 + j + 8 * hi;
    Obuf[obase + (size_t)n * C_ + lo] = o0[j] * inv;
    Obuf[obase + (size_t)n * C_ + 16 + lo] = o1[j] * inv;
  }

  // ---- per-block raw score sum: deterministic private slot ----
  ssum += __shfl_xor(ssum, 16);
  ssum += __shfl_xor(ssum, 8);
  ssum += __shfl_xor(ssum, 4);
  ssum += __shfl_xor(ssum, 2);
  ssum += __shfl_xor(ssum, 1);
  if (lane == 0) partial[(size_t)pair * NBLK + h * NTILE + blockIdx.x] = ssum;
}

// ---------------------------------------------------------------------------
// Deterministic partial-sum reduction, then pooled -> attn_r -> collapsed
// pair weights w[a][e].
// ---------------------------------------------------------------------------
__global__ __launch_bounds__(32) void weights_kernel(const float* __restrict__ partial,
                                                     float* __restrict__ w) {
  __shared__ float sumS[NPAIR];
  const int t = threadIdx.x;
  if (t < NPAIR) {
    float s = 0.0f;
    for (int i = 0; i < NBLK; ++i) s += partial[(size_t)t * NBLK + i];  // fixed order
    sumS[t] = s;
  }
  __syncthreads();
  if (t != 0) return;

  const int trace[12][4] = {{0, 1, 2, 3}, {0, 2, 3, 1}, {0, 3, 1, 2}, {1, 2, 0, 3},
                            {1, 0, 3, 2}, {1, 3, 2, 0}, {2, 3, 0, 1}, {2, 0, 1, 3},
                            {2, 1, 3, 0}, {3, 1, 0, 2}, {3, 0, 2, 1}, {3, 2, 1, 0}};
  const float inv_cnt = 1.0f / ((float)A_ * H_ * N_ * N_);
  float pos[12];
  float tot = 0.0f;
  for (int r = 0; r < 12; ++r) {
    float pooled = 0.0f;
    for (int a = 0; a < A_; ++a) pooled += sumS[a * 4 + trace[r][a]];
    pooled *= inv_cnt;
    pos[r] = pooled * pooled;
    tot += pos[r];
  }
  const float invtot = 1.0f / tot;
  for (int a = 0; a < A_; ++a)
    for (int e = 0; e < A_; ++e) {
      float acc = 0.0f;
      for (int r = 0; r < 12; ++r)
        if (trace[r][a] == e) acc += pos[r] * invtot;
      w[a * 4 + e] = acc;
    }
}

// ---------------------------------------------------------------------------
// hidden[a] = sum_e w[a,e] * O[a,e]; write out[B,A,N,D] (head-interleaved).
// blockIdx.y = a, so all index math is shifts/masks (no integer division).
// ---------------------------------------------------------------------------
__global__ __launch_bounds__(256) void combine_kernel(const float* __restrict__ Obuf,
                                                      const float* __restrict__ w,
                                                      float* __restrict__ out) {
  const int a = blockIdx.y;
  const int rem = blockIdx.x * blockDim.x + threadIdx.x;  // < N_*D_ = 98304
  const int n = rem >> 8;   // D_ = 256
  const int d = rem & 255;
  const int h = d >> 5;
  const int c = d & 31;
  float acc = 0.0f;
#pragma unroll
  for (int e = 0; e < A_; ++e) {
    const size_t oi = ((size_t)((a * 4 + e) * H_ + h)) * N_ * C_ + (size_t)n * C_ + c;
    acc += w[a * 4 + e] * Obuf[oi];
  }
  out[(size_t)a * N_ * D_ + rem] = acc;
}

// ---------------------------------------------------------------------------
extern "C" void kernel_launch(void* const* d_in, const int* in_sizes, int n_in,
                              void* d_out, int out_size, void* d_ws, size_t ws_size,
                              hipStream_t stream) {
  const float* input_q = (const float*)d_in[0];
  const float* input_k = (const float*)d_in[1];
  const float* input_v = (const float*)d_in[2];
  const float* Wq = (const float*)d_in[3];
  const float* bq = (const float*)d_in[4];
  const float* Wk = (const float*)d_in[5];
  const float* bk = (const float*)d_in[6];
  const float* Wv = (const float*)d_in[7];
  const float* bv = (const float*)d_in[8];
  float* out = (float*)d_out;

  float* ws = (float*)d_ws;
  float* qp = ws + WS_QP;
  float* kp = ws + WS_KP;
  float* vp = ws + WS_VP;
  float* Obuf = ws + WS_O;
  float* part = ws + WS_PART;
  float* wpair = ws + WS_W;

  dim3 pgrid((A_ * N_) / 16, D_ / 16);  // 96 x 16 tiles
  proj_kernel<<<pgrid, 32, 0, stream>>>(input_q, Wq, bq, qp);
  proj_kernel<<<pgrid, 32, 0, stream>>>(input_k, Wk, bk, kp);
  proj_kernel<<<pgrid, 32, 0, stream>>>(input_v, Wv, bv, vp);

  dim3 agrid(NTILE, H_, NPAIR);  // 24 x 8 x 16 one-wave blocks
  attn_kernel<<<agrid, 32, 0, stream>>>(qp, kp, vp, Obuf, part);

  weights_kernel<<<1, 32, 0, stream>>>(part, wpair);

  dim3 cgrid((N_ * D_) / 256, A_);  // 384 x 4
  combine_kernel<<<cgrid, 256, 0, stream>>>(Obuf, wpair, out);
}